// PyGT_TGCN_44787918963323
// MI455X (gfx1250) — compile-verified
//
#include <hip/hip_runtime.h>
#include <math.h>

typedef __attribute__((ext_vector_type(2))) float v2f;
typedef __attribute__((ext_vector_type(8))) float v8f;

#define F_IN  16
#define HDIM  32
#define F_OUT 8

__device__ __forceinline__ v8f wmma4(v2f a, v2f b, v8f c) {
    // D(16x16,f32) = A(16x4,f32) x B(4x16,f32) + C ; full fp32 precision
    return __builtin_amdgcn_wmma_f32_16x16x4_f32(false, a, false, b, (short)0, c, false, false);
}

__device__ __forceinline__ float sigmoidf_(float x) {
    return 1.0f / (1.0f + __expf(-x));
}

// ---------------------------------------------------------------- utilities
__global__ void k_zero(float* __restrict__ p, int n) {
    int i = blockIdx.x * blockDim.x + threadIdx.x;
    if (i < n) p[i] = 0.0f;
}

__global__ void k_deg(const int* __restrict__ dst, const float* __restrict__ ew,
                      float* __restrict__ deg, int ne) {
    int e = blockIdx.x * blockDim.x + threadIdx.x;
    if (e < ne) atomicAdd(&deg[dst[e]], ew[e]);
}

__global__ void k_dis(float* __restrict__ deg, int n) {
    int i = blockIdx.x * blockDim.x + threadIdx.x;
    if (i < n) {
        float d = deg[i];
        deg[i] = (d > 0.0f) ? rsqrtf(fmaxf(d, 1e-12f)) : 0.0f;
    }
}

// ---------------------------------------------------------------- XW = x @ [Wz|Wr|Wh]  -> [N,96]
__global__ void k_xw(const float* __restrict__ nf,
                     const float* __restrict__ Wz, const float* __restrict__ Wr,
                     const float* __restrict__ Wh,
                     float* __restrict__ xw, int nodes) {
    int wid  = (blockIdx.x * blockDim.x + threadIdx.x) >> 5;
    int lane = threadIdx.x & 31;
    int mt = wid / 6;          // 16-row node tile
    int ct = wid % 6;          // 16-col tile of the 96 output cols
    if (mt * 16 >= nodes) return;
    int r0    = mt << 4;
    int row16 = lane & 15;
    int koff  = (lane >> 4) << 1;   // 0 or 2
    int mbase = (lane >> 4) << 3;   // 0 or 8
    int gate  = ct >> 1;
    int lc    = ((ct & 1) << 4) + row16;   // local col within this gate's W (0..31)
    const float* W    = (gate == 0) ? Wz : (gate == 1) ? Wr : Wh;
    const float* arow = nf + (r0 + row16) * F_IN;

    v8f c = {};
    #pragma unroll
    for (int s = 0; s < 4; ++s) {
        int kb = s * 4 + koff;
        v2f a, b;
        a.x = arow[kb];
        a.y = arow[kb + 1];
        b.x = W[kb * HDIM + lc];
        b.y = W[(kb + 1) * HDIM + lc];
        c = wmma4(a, b, c);
    }
    int col = gate * HDIM + lc;     // 0..95
    #pragma unroll
    for (int v = 0; v < 8; ++v)
        xw[(r0 + mbase + v) * 96 + col] = c[v];
}

// ---------------------------------------------------------------- edge scatter (one wave per edge)
__global__ void k_scatter(const int* __restrict__ src, const int* __restrict__ dst,
                          const float* __restrict__ ew, const float* __restrict__ dis,
                          const float* __restrict__ xw, float* __restrict__ agg, int ne) {
    int t = blockIdx.x * blockDim.x + threadIdx.x;
    int e = t >> 5, lane = t & 31;
    if (e >= ne) return;
    int s = src[e], d = dst[e];
    float nrm = dis[s] * ew[e] * dis[d];
    const float* xs = xw + (size_t)s * 96;
    float*       ad = agg + (size_t)d * 96;
    #pragma unroll
    for (int k = 0; k < 3; ++k) {
        int ccol = lane + (k << 5);
        atomicAdd(ad + ccol, nrm * xs[ccol]);
    }
}

// ---------------------------------------------------------------- fused GRU + head MLP, 16 nodes/wave
__global__ void __launch_bounds__(32)
k_gru(const float* __restrict__ agg, const float* __restrict__ Hs,
      const float* __restrict__ bz, const float* __restrict__ br, const float* __restrict__ bh,
      const float* __restrict__ Lzw, const float* __restrict__ Lzb,
      const float* __restrict__ Lrw, const float* __restrict__ Lrb,
      const float* __restrict__ Lhw, const float* __restrict__ Lhb,
      const float* __restrict__ l1w, const float* __restrict__ l1b,
      const float* __restrict__ l2w, const float* __restrict__ l2b,
      float* __restrict__ oy2, float* __restrict__ oy1, float* __restrict__ oh) {
    __shared__ float zbuf[16 * 32];   // sigmoid(z) tile
    __shared__ float hrbuf[16 * 32];  // Hs * sigmoid(r) tile
    __shared__ float ybuf[16 * 32];   // relu(h) tile
    __shared__ float y2buf[16 * 16];  // y (after lin1) tile

    int mt    = blockIdx.x;
    int lane  = threadIdx.x & 31;
    int r0    = mt << 4;
    int row16 = lane & 15;
    int koff  = (lane >> 4) << 1;
    int mbase = (lane >> 4) << 3;

    const float* arow = agg + (r0 + row16) * 96;
    const float* hrow = Hs + (r0 + row16) * HDIM;

    // ---- z and r gates: [conv_g | Hs](16x64) @ L(64x32), two 16-col tiles ----
    #pragma unroll
    for (int cc = 0; cc < 2; ++cc) {
        int col = (cc << 4) + row16;
        v8f cz = {}, cr = {};
        #pragma unroll
        for (int s = 0; s < 16; ++s) {
            int kb = s * 4 + koff;          // kb even; kb,kb+1 on same side of 32
            v2f az, ar, bzv, brv;
            if (kb < HDIM) {
                az.x = arow[kb]      + bz[kb];
                az.y = arow[kb + 1]  + bz[kb + 1];
                ar.x = arow[32 + kb]     + br[kb];
                ar.y = arow[32 + kb + 1] + br[kb + 1];
            } else {
                az.x = hrow[kb - 32];
                az.y = hrow[kb - 31];
                ar = az;
            }
            bzv.x = Lzw[kb * HDIM + col];
            bzv.y = Lzw[(kb + 1) * HDIM + col];
            brv.x = Lrw[kb * HDIM + col];
            brv.y = Lrw[(kb + 1) * HDIM + col];
            cz = wmma4(az, bzv, cz);
            cr = wmma4(ar, brv, cr);
        }
        #pragma unroll
        for (int v = 0; v < 8; ++v) {
            int m = mbase + v;
            float zval = sigmoidf_(cz[v] + Lzb[col]);
            float rval = sigmoidf_(cr[v] + Lrb[col]);
            zbuf[m * 32 + col]  = zval;
            hrbuf[m * 32 + col] = rval * Hs[(r0 + m) * HDIM + col];
        }
    }
    __syncthreads();

    // ---- candidate gate: [conv_h | Hs*r](16x64) @ Lh(64x32), tanh, GRU combine ----
    #pragma unroll
    for (int cc = 0; cc < 2; ++cc) {
        int col = (cc << 4) + row16;
        v8f ch = {};
        #pragma unroll
        for (int s = 0; s < 16; ++s) {
            int kb = s * 4 + koff;
            v2f a, b;
            if (kb < HDIM) {
                a.x = arow[64 + kb]     + bh[kb];
                a.y = arow[64 + kb + 1] + bh[kb + 1];
            } else {
                a.x = hrbuf[row16 * 32 + kb - 32];
                a.y = hrbuf[row16 * 32 + kb - 31];
            }
            b.x = Lhw[kb * HDIM + col];
            b.y = Lhw[(kb + 1) * HDIM + col];
            ch = wmma4(a, b, ch);
        }
        #pragma unroll
        for (int v = 0; v < 8; ++v) {
            int m = mbase + v;
            int node = r0 + m;
            float ht = tanhf(ch[v] + Lhb[col]);
            float zv = zbuf[m * 32 + col];
            float hs = Hs[node * HDIM + col];
            float h  = zv * hs + (1.0f - zv) * ht;
            oh[node * HDIM + col] = h;
            ybuf[m * 32 + col] = fmaxf(h, 0.0f);
        }
    }
    __syncthreads();

    // ---- y = relu(h)(16x32) @ lin1(32x16) + b ----
    {
        int col = row16;
        v8f cy = {};
        #pragma unroll
        for (int s = 0; s < 8; ++s) {
            int kb = s * 4 + koff;
            v2f a, b;
            a.x = ybuf[row16 * 32 + kb];
            a.y = ybuf[row16 * 32 + kb + 1];
            b.x = l1w[kb * F_IN + col];
            b.y = l1w[(kb + 1) * F_IN + col];
            cy = wmma4(a, b, cy);
        }
        #pragma unroll
        for (int v = 0; v < 8; ++v) {
            int m = mbase + v;
            float yv = cy[v] + l1b[col];
            oy1[(r0 + m) * F_IN + col] = yv;
            y2buf[m * 16 + col] = yv;
        }
    }
    __syncthreads();

    // ---- y_out = y(16x16) @ lin2(16x8) + b  (B cols 8..15 padded with 0) ----
    {
        int col = row16;
        v8f co = {};
        #pragma unroll
        for (int s = 0; s < 4; ++s) {
            int kb = s * 4 + koff;
            v2f a, b;
            a.x = y2buf[row16 * 16 + kb];
            a.y = y2buf[row16 * 16 + kb + 1];
            b.x = (col < F_OUT) ? l2w[kb * F_OUT + col] : 0.0f;
            b.y = (col < F_OUT) ? l2w[(kb + 1) * F_OUT + col] : 0.0f;
            co = wmma4(a, b, co);
        }
        if (col < F_OUT) {
            #pragma unroll
            for (int v = 0; v < 8; ++v)
                oy2[(r0 + mbase + v) * F_OUT + col] = co[v] + l2b[col];
        }
    }
}

// ---------------------------------------------------------------- launcher
extern "C" void kernel_launch(void* const* d_in, const int* in_sizes, int n_in,
                              void* d_out, int out_size, void* d_ws, size_t ws_size,
                              hipStream_t stream) {
    const float* node_feat = (const float*)d_in[0];
    const int*   eidx      = (const int*)d_in[1];
    const float* ew        = (const float*)d_in[2];
    const float* Hs        = (const float*)d_in[3];
    const float* Wz  = (const float*)d_in[4];
    const float* bz  = (const float*)d_in[5];
    const float* Wr  = (const float*)d_in[6];
    const float* br  = (const float*)d_in[7];
    const float* Wh  = (const float*)d_in[8];
    const float* bh  = (const float*)d_in[9];
    const float* Lzw = (const float*)d_in[10];
    const float* Lzb = (const float*)d_in[11];
    const float* Lrw = (const float*)d_in[12];
    const float* Lrb = (const float*)d_in[13];
    const float* Lhw = (const float*)d_in[14];
    const float* Lhb = (const float*)d_in[15];
    const float* l1w = (const float*)d_in[16];
    const float* l1b = (const float*)d_in[17];
    const float* l2w = (const float*)d_in[18];
    const float* l2b = (const float*)d_in[19];

    const int nodes = in_sizes[0] / F_IN;   // 100000 (divisible by 16)
    const int nedge = in_sizes[2];          // 1600000
    const int* src = eidx;
    const int* dst = eidx + nedge;

    // workspace: dis[N] | xw[N,96] | agg[N,96]  (~77 MB)
    float* dis = (float*)d_ws;
    float* xw  = dis + nodes;
    float* agg = xw + (size_t)nodes * 96;

    // d_out = y_out[N,8] | y[N,16] | h[N,32]
    float* oy2 = (float*)d_out;
    float* oy1 = oy2 + (size_t)nodes * F_OUT;
    float* oh  = oy1 + (size_t)nodes * F_IN;

    k_zero<<<(nodes + 255) / 256, 256, 0, stream>>>(dis, nodes);
    {
        int m = nodes * 96;
        k_zero<<<(m + 255) / 256, 256, 0, stream>>>(agg, m);
    }
    k_deg<<<(nedge + 255) / 256, 256, 0, stream>>>(dst, ew, dis, nedge);
    k_dis<<<(nodes + 255) / 256, 256, 0, stream>>>(dis, nodes);
    {
        int waves   = ((nodes + 15) / 16) * 6;
        int threads = waves * 32;
        k_xw<<<(threads + 255) / 256, 256, 0, stream>>>(node_feat, Wz, Wr, Wh, xw, nodes);
    }
    {
        long long threads = (long long)nedge * 32;
        k_scatter<<<(int)((threads + 255) / 256), 256, 0, stream>>>(src, dst, ew, dis, xw, agg, nedge);
    }
    k_gru<<<(nodes + 15) / 16, 32, 0, stream>>>(agg, Hs, bz, br, bh,
                                                Lzw, Lzb, Lrw, Lrb, Lhw, Lhb,
                                                l1w, l1b, l2w, l2b, oy2, oy1, oh);
}